// Line_18408229831259
// MI455X (gfx1250) — compile-verified
//
#include <hip/hip_runtime.h>
#include <hip/hip_bf16.h>
#include <math.h>

// ---------------------------------------------------------------------------
// Sizes (fixed by the reference)
// ---------------------------------------------------------------------------
#define BATCH 4096
#define DIM   128
#define LBL   50
#define NEGK  5
#define LD3   64            // padded row stride for the [B,50] logits

#define NT_PER_WAVE 8                 // N-tiles (16 cols each) per wave
#define N_POS_WAVES (2 * 256 * (256 / NT_PER_WAVE))  // 2 embs x 256 m-tiles x 32 n-groups = 16384
#define N_NEG       (2 * BATCH * NEGK)               // 40960

typedef __attribute__((ext_vector_type(2))) float v2f;
typedef __attribute__((ext_vector_type(8))) float v8f;

__device__ __forceinline__ float log_sigmoid(float x) {
    // log(sigmoid(x)) = min(x,0) - log1p(exp(-|x|))
    return fminf(x, 0.0f) - log1pf(expf(-fabsf(x)));
}

// ---------------------------------------------------------------------------
// 1) Gather: vi = U[s], vou = U[t], voc = C[t]; also build h0 = [vi | vou]
//    grid = BATCH blocks x 128 threads
// ---------------------------------------------------------------------------
__global__ void gather_kernel(const int* __restrict__ s, const int* __restrict__ t,
                              const float* __restrict__ u_emb, const float* __restrict__ ctx_emb,
                              float* __restrict__ vi, float* __restrict__ vou,
                              float* __restrict__ voc, float* __restrict__ h0) {
    const int row = blockIdx.x;
    const int j   = threadIdx.x;
    const size_t si = (size_t)s[row] * DIM;
    const size_t ti = (size_t)t[row] * DIM;
    const float a  = u_emb[si + j];
    const float bu = u_emb[ti + j];
    const float bc = ctx_emb[ti + j];
    vi [(size_t)row * DIM + j] = a;
    vou[(size_t)row * DIM + j] = bu;
    voc[(size_t)row * DIM + j] = bc;
    h0 [(size_t)row * (2 * DIM) + j]        = a;
    h0 [(size_t)row * (2 * DIM) + DIM + j]  = bu;
}

// ---------------------------------------------------------------------------
// 2) Negative-sample dots: one wave per (emb, b, k); p_neg[i] = ls(-vi[b].vn)
//    grid = N_NEG/8 blocks x 256 threads
// ---------------------------------------------------------------------------
__global__ void neg_kernel(const float* __restrict__ vi,
                           const float* __restrict__ u_emb, const float* __restrict__ ctx_emb,
                           const int* __restrict__ ng, float* __restrict__ p_neg) {
    const int wave = blockIdx.x * 8 + (threadIdx.x >> 5);
    const int lane = threadIdx.x & 31;
    const int emb  = wave / (BATCH * NEGK);
    const int rem  = wave % (BATCH * NEGK);
    const int b    = rem / NEGK;
    const int k    = rem % NEGK;
    const float* E = emb ? ctx_emb : u_emb;
    const size_t node = (size_t)ng[b * NEGK + k];
    const float* vn = E + node * DIM;
    const float* v  = vi + (size_t)b * DIM;
    float d = 0.0f;
    #pragma unroll
    for (int j = lane; j < DIM; j += 32) d = fmaf(v[j], vn[j], d);
    #pragma unroll
    for (int off = 16; off; off >>= 1) d += __shfl_down(d, off, 32);
    if (lane == 0) p_neg[wave] = log_sigmoid(-d);
}

// ---------------------------------------------------------------------------
// 3) All-pairs matmul via V_WMMA_F32_16X16X4_F32, register-blocked.
//    One wave -> one 16x128 strip: A fragment (full K=128) loaded ONCE into
//    64 VGPRs and reused across 8 N-tiles. B fragments are batch-loaded in
//    two 16-step halves per tile, so loads pipeline against the WMMA chain
//    while keeping the live-register peak spill-free.
//    __launch_bounds__(256, 1) lifts the occupancy-driven VGPR cap so the
//    A fragments stay register-resident (no scratch spills).
//    fp32 A 16x4 frag:  lanes 0-15: M=l, K={k0,k0+1}; lanes 16-31: K={k0+2,k0+3}
//    fp32 B 4x16 frag (B[k][n]=vo[n][k]): identical per-lane pattern on vo rows.
//    grid = N_POS_WAVES/8 blocks x 256 threads (EXEC all-ones, no divergence)
// ---------------------------------------------------------------------------
__global__ void __launch_bounds__(256, 1)
pos_wmma_kernel(const float* __restrict__ vi,
                const float* __restrict__ vou, const float* __restrict__ voc,
                float* __restrict__ p_pos) {
    const int wave = blockIdx.x * 8 + (threadIdx.x >> 5);
    const int lane = threadIdx.x & 31;
    const int emb  = wave >> 13;              // 0: u_emb, 1: ctx_emb   (wave / 8192)
    const int rem  = wave & 8191;
    const int mt   = rem >> 5;                // 256 m-tiles
    const int ngp  = rem & 31;                // 32 n-groups of NT_PER_WAVE tiles
    const int tm   = mt << 4;
    const int tn0  = ngp << 7;                // 8 tiles * 16 cols = 128 cols per group
    const float* Bmat = emb ? voc : vou;

    const int grp = lane >> 4;                // 0 -> K{0,1}, 1 -> K{2,3}
    const int l   = lane & 15;

    // A fragments for the full K=128: resident for the whole strip.
    const float* arow = vi + (size_t)(tm + l) * DIM + 2 * grp;
    v2f a[32];
    #pragma unroll
    for (int st = 0; st < 32; ++st) a[st] = *(const v2f*)(arow + 4 * st);

    float ssum = 0.0f;
    #pragma unroll
    for (int nt = 0; nt < NT_PER_WAVE; ++nt) {
        const float* brow = Bmat + (size_t)(tn0 + nt * 16 + l) * DIM + 2 * grp;

        v8f acc = {};
        // First half: batch 16 B loads, then 16 WMMAs.
        v2f b0[16];
        #pragma unroll
        for (int st = 0; st < 16; ++st) b0[st] = *(const v2f*)(brow + 4 * st);
        // Second half loads can be issued while the first 16 WMMAs run.
        v2f b1[16];
        #pragma unroll
        for (int st = 0; st < 16; ++st) b1[st] = *(const v2f*)(brow + 64 + 4 * st);

        #pragma unroll
        for (int st = 0; st < 16; ++st) {
            acc = __builtin_amdgcn_wmma_f32_16x16x4_f32(
                false, a[st], false, b0[st], (short)0, acc, false, false);
        }
        #pragma unroll
        for (int st = 0; st < 16; ++st) {
            acc = __builtin_amdgcn_wmma_f32_16x16x4_f32(
                false, a[16 + st], false, b1[st], (short)0, acc, false, false);
        }

        #pragma unroll
        for (int r = 0; r < 8; ++r) ssum += log_sigmoid(acc[r]);
    }

    #pragma unroll
    for (int off = 16; off; off >>= 1) ssum += __shfl_down(ssum, off, 32);
    if (lane == 0) p_pos[wave] = ssum;
}

// ---------------------------------------------------------------------------
// 4) Generic out = in @ W + bias   (W row-major [K,N])
//    grid = BATCH blocks x ldout threads; block shares one input row (scalar
//    broadcast loads), W reads coalesced across columns.
// ---------------------------------------------------------------------------
__global__ void matmul_bias_kernel(const float* __restrict__ in, const float* __restrict__ W,
                                   const float* __restrict__ bias, float* __restrict__ out,
                                   int K, int N, int ldin, int ldout) {
    const int row = blockIdx.x;
    const int col = threadIdx.x;
    if (col >= N) return;
    const float* ir = in + (size_t)row * ldin;
    float acc = bias[col];
    for (int k = 0; k < K; ++k) acc = fmaf(ir[k], W[(size_t)k * N + col], acc);
    out[(size_t)row * ldout + col] = acc;
}

// ---------------------------------------------------------------------------
// 5) Train-mode BatchNorm column stats (biased variance). One block / column.
// ---------------------------------------------------------------------------
__global__ void bn_stats_kernel(const float* __restrict__ z, float* __restrict__ mean,
                                float* __restrict__ inv, int M, int ld, float eps) {
    const int c   = blockIdx.x;
    const int tid = threadIdx.x;
    float s1 = 0.0f, s2 = 0.0f;
    for (int r = tid; r < M; r += 256) {
        const float v = z[(size_t)r * ld + c];
        s1 += v;
        s2 = fmaf(v, v, s2);
    }
    __shared__ float a[256], b[256];
    a[tid] = s1; b[tid] = s2;
    __syncthreads();
    for (int off = 128; off; off >>= 1) {
        if (tid < off) { a[tid] += a[tid + off]; b[tid] += b[tid + off]; }
        __syncthreads();
    }
    if (tid == 0) {
        const float m   = a[0] / (float)M;
        const float var = b[0] / (float)M - m * m;
        mean[c] = m;
        inv[c]  = rsqrtf(var + eps);
    }
}

// ---------------------------------------------------------------------------
// 6) In-place BN apply + ReLU
// ---------------------------------------------------------------------------
__global__ void bn_relu_kernel(float* __restrict__ z, const float* __restrict__ mean,
                               const float* __restrict__ inv, const float* __restrict__ g,
                               const float* __restrict__ be, int N, int ld) {
    const int row = blockIdx.x;
    const int col = threadIdx.x;
    if (col >= N) return;
    const size_t i = (size_t)row * ld + col;
    const float v = (z[i] - mean[col]) * inv[col] * g[col] + be[col];
    z[i] = fmaxf(v, 0.0f);
}

// ---------------------------------------------------------------------------
// 7) log-softmax CE partials: one wave per row, p_ce[row] = logp[row, y[row]]
// ---------------------------------------------------------------------------
__global__ void ce_kernel(const float* __restrict__ z3, const int* __restrict__ y,
                          float* __restrict__ p_ce) {
    const int row  = blockIdx.x * 8 + (threadIdx.x >> 5);
    const int lane = threadIdx.x & 31;
    const float* r = z3 + (size_t)row * LD3;
    const float v0 = (lane < LBL)        ? r[lane]      : -INFINITY;
    const float v1 = ((lane + 32) < LBL) ? r[lane + 32] : -INFINITY;
    float mx = fmaxf(v0, v1);
    #pragma unroll
    for (int off = 16; off; off >>= 1) mx = fmaxf(mx, __shfl_xor(mx, off, 32));
    float se = ((lane < LBL) ? expf(v0 - mx) : 0.0f) +
               (((lane + 32) < LBL) ? expf(v1 - mx) : 0.0f);
    #pragma unroll
    for (int off = 16; off; off >>= 1) se += __shfl_xor(se, off, 32);
    if (lane == 0) {
        const int yy = y[row];
        p_ce[row] = r[yy] - mx - logf(se);
    }
}

// ---------------------------------------------------------------------------
// 8) Deterministic final reduction (fixed strided order + fixed LDS tree):
//    out = -(pos/B^2) - (neg/B) - (ce/B)
// ---------------------------------------------------------------------------
__global__ void finalize_kernel(const float* __restrict__ p_pos, const float* __restrict__ p_neg,
                                const float* __restrict__ p_ce, float* __restrict__ out) {
    const int tid = threadIdx.x;
    __shared__ float sa[256], sb[256], sc[256];
    float a = 0.0f, b = 0.0f, c = 0.0f;
    for (int i = tid; i < N_POS_WAVES; i += 256) a += p_pos[i];
    for (int i = tid; i < N_NEG;       i += 256) b += p_neg[i];
    for (int i = tid; i < BATCH;       i += 256) c += p_ce[i];
    sa[tid] = a; sb[tid] = b; sc[tid] = c;
    __syncthreads();
    for (int off = 128; off; off >>= 1) {
        if (tid < off) { sa[tid] += sa[tid + off]; sb[tid] += sb[tid + off]; sc[tid] += sc[tid + off]; }
        __syncthreads();
    }
    if (tid == 0) {
        const double invB  = 1.0 / (double)BATCH;
        const double invB2 = invB * invB;
        out[0] = (float)(-(double)sa[0] * invB2 - (double)sb[0] * invB - (double)sc[0] * invB);
    }
}

// ---------------------------------------------------------------------------
// Launch
// ---------------------------------------------------------------------------
extern "C" void kernel_launch(void* const* d_in, const int* in_sizes, int n_in,
                              void* d_out, int out_size, void* d_ws, size_t ws_size,
                              hipStream_t stream) {
    const int*   s       = (const int*)  d_in[0];
    const int*   t       = (const int*)  d_in[1];
    const int*   ng      = (const int*)  d_in[2];
    const int*   y       = (const int*)  d_in[3];
    const float* u_emb   = (const float*)d_in[4];
    const float* ctx_emb = (const float*)d_in[5];
    const float* W1      = (const float*)d_in[6];
    const float* b1      = (const float*)d_in[7];
    const float* g1      = (const float*)d_in[8];
    const float* be1     = (const float*)d_in[9];
    const float* W2      = (const float*)d_in[10];
    const float* b2      = (const float*)d_in[11];
    const float* g2      = (const float*)d_in[12];
    const float* be2     = (const float*)d_in[13];
    const float* W3      = (const float*)d_in[14];
    const float* b3      = (const float*)d_in[15];
    const float* g3      = (const float*)d_in[16];
    const float* be3     = (const float*)d_in[17];
    float* out = (float*)d_out;

    // Workspace carve-up (floats)
    float* ws = (float*)d_ws;
    size_t off = 0;
    float* vi    = ws + off; off += (size_t)BATCH * DIM;        // 524288
    float* vou   = ws + off; off += (size_t)BATCH * DIM;
    float* voc   = ws + off; off += (size_t)BATCH * DIM;
    float* h0    = ws + off; off += (size_t)BATCH * 2 * DIM;    // [B,256]
    float* z1    = ws + off; off += (size_t)BATCH * 2 * DIM;    // [B,256]
    float* z2    = ws + off; off += (size_t)BATCH * DIM;        // [B,128]
    float* z3    = ws + off; off += (size_t)BATCH * LD3;        // [B,64] (50 used)
    float* meanb = ws + off; off += 256;
    float* invb  = ws + off; off += 256;
    float* p_pos = ws + off; off += N_POS_WAVES;                // 16384
    float* p_neg = ws + off; off += N_NEG;                      // 40960
    float* p_ce  = ws + off; off += BATCH;                      // 4096
    (void)ws_size; (void)in_sizes; (void)n_in; (void)out_size;

    // Gather embedding rows + build concat input
    gather_kernel<<<BATCH, DIM, 0, stream>>>(s, t, u_emb, ctx_emb, vi, vou, voc, h0);

    // Negative-sample log-sigmoid partials
    neg_kernel<<<N_NEG / 8, 256, 0, stream>>>(vi, u_emb, ctx_emb, ng, p_neg);

    // All-pairs WMMA strips + log-sigmoid partials (both embeddings)
    pos_wmma_kernel<<<N_POS_WAVES / 8, 256, 0, stream>>>(vi, vou, voc, p_pos);

    // MLP layer 1: [B,256] @ [256,256] + BN(eps=2D) + ReLU
    matmul_bias_kernel<<<BATCH, 256, 0, stream>>>(h0, W1, b1, z1, 256, 256, 256, 256);
    bn_stats_kernel<<<256, 256, 0, stream>>>(z1, meanb, invb, BATCH, 256, 2.0f * DIM);
    bn_relu_kernel<<<BATCH, 256, 0, stream>>>(z1, meanb, invb, g1, be1, 256, 256);

    // MLP layer 2: [B,256] @ [256,128] + BN(eps=D) + ReLU
    matmul_bias_kernel<<<BATCH, 128, 0, stream>>>(z1, W2, b2, z2, 256, 128, 256, 128);
    bn_stats_kernel<<<128, 256, 0, stream>>>(z2, meanb, invb, BATCH, 128, (float)DIM);
    bn_relu_kernel<<<BATCH, 128, 0, stream>>>(z2, meanb, invb, g2, be2, 128, 128);

    // MLP layer 3: [B,128] @ [128,50] + BN(eps=D) + ReLU
    matmul_bias_kernel<<<BATCH, LD3, 0, stream>>>(z2, W3, b3, z3, 128, LBL, 128, LD3);
    bn_stats_kernel<<<LBL, 256, 0, stream>>>(z3, meanb, invb, BATCH, LD3, (float)DIM);
    bn_relu_kernel<<<BATCH, LD3, 0, stream>>>(z3, meanb, invb, g3, be3, LBL, LD3);

    // Cross entropy partials + deterministic combine
    ce_kernel<<<BATCH / 8, 256, 0, stream>>>(z3, y, p_ce);
    finalize_kernel<<<1, 256, 0, stream>>>(p_pos, p_neg, p_ce, out);
}